// DeepFit_53188874993720
// MI455X (gfx1250) — compile-verified
//
#include <hip/hip_runtime.h>

typedef float v2f __attribute__((ext_vector_type(2)));
typedef float v4f __attribute__((ext_vector_type(4)));
typedef float v8f __attribute__((ext_vector_type(8)));

#define NPTS  1024
#define BLK   256
#define NWAVE 8            // BLK / 32 (wave32)
#define SPW   (NPTS/NWAVE) // 128 samples per wave
#define STEPS (SPW/4)      // 32 WMMA steps of K=4

// feat = x^px * y^py, branchless
__device__ __forceinline__ float featpow(float x, float y, int px, int py) {
    float r = (px >= 1) ? x : 1.0f;
    r *= (px >= 2) ? x : 1.0f;
    r *= (px >= 3) ? x : 1.0f;
    r *= (py >= 1) ? y : 1.0f;
    r *= (py >= 2) ? y : 1.0f;
    r *= (py >= 3) ? y : 1.0f;
    return r;
}

__global__ __launch_bounds__(BLK) void wjet_kernel(
    const float* __restrict__ points, const float* __restrict__ weights,
    float* __restrict__ out_beta, float* __restrict__ out_nest,
    float* __restrict__ out_nn)
{
    __shared__ float xs[NPTS], ys[NPTS], zs[NPTS], ws[NPTS];
    __shared__ float red[BLK * 3];
    __shared__ float accred[NWAVE * 256];
    __shared__ float mat[256];        // 16x16: XtX rows/cols 0..9, XtY in col 10
    __shared__ float sh_beta[10];
    __shared__ float sh_scal[2];      // h, use_w

    const int b   = blockIdx.x;
    const int tid = threadIdx.x;

    const float* gx = points + (size_t)b * 3 * NPTS;
    const float* gy = gx + NPTS;
    const float* gz = gx + 2 * NPTS;
    const float* gw = weights + (size_t)b * NPTS;

    // ---- Phase 1: stage batch into LDS (b128) + partial reductions ----
    // NPTS/4 == BLK, so each thread handles exactly one float4 per array.
    {
        const v4f xv = __builtin_nontemporal_load((const v4f*)gx + tid);
        const v4f yv = __builtin_nontemporal_load((const v4f*)gy + tid);
        const v4f zv = __builtin_nontemporal_load((const v4f*)gz + tid);
        const v4f wv = __builtin_nontemporal_load((const v4f*)gw + tid);
        *(v4f*)&xs[4 * tid] = xv;
        *(v4f*)&ys[4 * tid] = yv;
        *(v4f*)&zs[4 * tid] = zv;
        *(v4f*)&ws[4 * tid] = wv;
        float sax = fabsf(xv[0]) + fabsf(xv[1]) + fabsf(xv[2]) + fabsf(xv[3]);
        float say = fabsf(yv[0]) + fabsf(yv[1]) + fabsf(yv[2]) + fabsf(yv[3]);
        float cnt = 0.f;
#pragma unroll
        for (int j = 0; j < 4; ++j) cnt += (wv[j] > 0.001f) ? 1.0f : 0.0f;
        red[tid] = sax; red[BLK + tid] = say; red[2 * BLK + tid] = cnt;
    }
    __syncthreads();
    if (tid < 32) {
        float a = 0.f, bb = 0.f, c = 0.f;
        for (int j = tid; j < BLK; j += 32) { a += red[j]; bb += red[BLK + j]; c += red[2 * BLK + j]; }
        red[tid] = a; red[BLK + tid] = bb; red[2 * BLK + tid] = c;
    }
    __syncthreads();
    if (tid == 0) {
        float a = 0.f, bb = 0.f, c = 0.f;
        for (int j = 0; j < 32; ++j) { a += red[j]; bb += red[BLK + j]; c += red[2 * BLK + j]; }
        float h = (a + bb) / (2.0f * NPTS);
        if (fabsf(h) < 0.0001f) h = 0.1f;
        sh_scal[0] = h;
        sh_scal[1] = (c > 18.0f) ? 1.0f : 0.0f;
    }
    __syncthreads();
    const float h     = sh_scal[0];
    const bool  use_w = sh_scal[1] > 0.5f;
    const float inv_h = 1.0f / h;

    // Pre-pass: scale coords, bake weight selection + w*z into LDS so the
    // WMMA loop is pure load->valu->wmma with no conditional consumers.
    {
        v4f xv = *(v4f*)&xs[4 * tid];
        v4f yv = *(v4f*)&ys[4 * tid];
        v4f zv = *(v4f*)&zs[4 * tid];
        v4f wv = *(v4f*)&ws[4 * tid];
#pragma unroll
        for (int j = 0; j < 4; ++j) {
            const float we = use_w ? wv[j] : 1.0f;
            wv[j] = we;
            zv[j] = zv[j] * we;   // zs now holds w_eff * z
            xv[j] *= inv_h;
            yv[j] *= inv_h;
        }
        *(v4f*)&xs[4 * tid] = xv;
        *(v4f*)&ys[4 * tid] = yv;
        *(v4f*)&zs[4 * tid] = zv;
        *(v4f*)&ws[4 * tid] = wv;
    }
    __syncthreads();

    // ---- Phase 2: XtX / XtY via V_WMMA_F32_16X16X4_F32 ----
    const int wave = tid >> 5;
    const int lane = tid & 31;
    const int g    = (lane >> 4) & 1;  // K half: lanes 0-15 -> K0,K1 ; lanes 16-31 -> K2,K3
    const int i    = lane & 15;        // feature row (A) / column (B)

    // packed exponent tables for features [x,y,x2,y2,xy,x3,y3,x2y,xy2,1]
    const unsigned long long pxt = 0x0120310201ull;
    const unsigned long long pyt = 0x0213012010ull;
    const int px = (int)((pxt >> (4 * i)) & 0xF);
    const int py = (int)((pyt >> (4 * i)) & 0xF);
    const float selZ = (i == 10) ? 1.0f : 0.0f;   // column 10 carries w*z
    const float selW = 1.0f - selZ;

    v8f acc = {0.f, 0.f, 0.f, 0.f, 0.f, 0.f, 0.f, 0.f};

    for (int s = 0; s < STEPS; ++s) {
        const int k0 = wave * SPW + s * 4 + 2 * g;   // even -> 8B-aligned LDS
        const v2f xv = *(const v2f*)&xs[k0];
        const v2f yv = *(const v2f*)&ys[k0];
        const v2f zv = *(const v2f*)&zs[k0];         // = w_eff * z
        const v2f wv = *(const v2f*)&ws[k0];         // = w_eff

        const float f0 = featpow(xv[0], yv[0], px, py);
        const float f1 = featpow(xv[1], yv[1], px, py);

        v2f Av; Av[0] = f0; Av[1] = f1;              // unweighted features
        v2f Bv;                                      // weighted features / col10 = w*z
        Bv[0] = fmaf(wv[0] * f0, selW, zv[0] * selZ);
        Bv[1] = fmaf(wv[1] * f1, selW, zv[1] * selZ);

        acc = __builtin_amdgcn_wmma_f32_16x16x4_f32(
            /*neg_a=*/false, Av, /*neg_b=*/false, Bv,
            /*c_mod=*/(short)0, acc, /*reuse_a=*/false, /*reuse_b=*/false);
    }

    // ---- Phase 3: reduce 8 wave accumulators ----
    {
        v4f lo; lo[0] = acc[0]; lo[1] = acc[1]; lo[2] = acc[2]; lo[3] = acc[3];
        v4f hi; hi[0] = acc[4]; hi[1] = acc[5]; hi[2] = acc[6]; hi[3] = acc[7];
        *(v4f*)&accred[wave * 256 + lane * 8 + 0] = lo;
        *(v4f*)&accred[wave * 256 + lane * 8 + 4] = hi;
    }
    __syncthreads();
    {
        float ssum = 0.f;
#pragma unroll
        for (int wv = 0; wv < NWAVE; ++wv) ssum += accred[wv * 256 + tid];
        // map slot -> (m, c) per 16x16 f32 C/D layout
        const int L = tid >> 3, j = tid & 7;
        const int m = (L < 16) ? j : (j + 8);
        const int c = L & 15;
        mat[m * 16 + c] = ssum;
    }
    __syncthreads();

    // ---- Phase 4: Cholesky 10x10 + solves + beta outputs (thread 0) ----
    if (tid == 0) {
        for (int j = 0; j < 10; ++j) {
            float s = mat[j * 16 + j];
            for (int k = 0; k < j; ++k) { float l = mat[j * 16 + k]; s -= l * l; }
            const float d = sqrtf(s);
            mat[j * 16 + j] = d;
            const float invd = 1.0f / d;
            for (int r = j + 1; r < 10; ++r) {
                float s2 = mat[r * 16 + j];
                for (int k = 0; k < j; ++k) s2 -= mat[r * 16 + k] * mat[j * 16 + k];
                mat[r * 16 + j] = s2 * invd;
            }
        }
        float u[10], bl[10];
        for (int j = 0; j < 10; ++j) {                 // L u = XtY
            float s = mat[j * 16 + 10];
            for (int k = 0; k < j; ++k) s -= mat[j * 16 + k] * u[k];
            u[j] = s / mat[j * 16 + j];
        }
        for (int j = 9; j >= 0; --j) {                 // L^T beta = u
            float s = u[j];
            for (int k = j + 1; k < 10; ++k) s -= mat[k * 16 + j] * bl[k];
            bl[j] = s / mat[j * 16 + j];
        }
        const float h2 = h * h, h3 = h2 * h;
        const float dinv[10] = {1.f / h, 1.f / h, 1.f / h2, 1.f / h2, 1.f / h2,
                                1.f / h3, 1.f / h3, 1.f / h3, 1.f / h3, 1.f};
        for (int j = 0; j < 10; ++j) {
            const float v = bl[j] * dinv[j];
            sh_beta[j] = v;
            out_beta[(size_t)b * 10 + j] = v;
        }
        const float n0 = -sh_beta[0], n1 = -sh_beta[1];
        const float inv = 1.0f / sqrtf(n0 * n0 + n1 * n1 + 1.0f);
        out_nest[(size_t)b * 3 + 0] = n0 * inv;
        out_nest[(size_t)b * 3 + 1] = n1 * inv;
        out_nest[(size_t)b * 3 + 2] = inv;
    }
    __syncthreads();

    // ---- Phase 5: per-point neighbor normals, packed b128 NT stores ----
    {
        const float b0 = sh_beta[0], b1 = sh_beta[1], b2 = sh_beta[2], b3 = sh_beta[3];
        const float b4 = sh_beta[4], b5 = sh_beta[5], b6 = sh_beta[6], b7 = sh_beta[7];
        const float b8 = sh_beta[8];
        const int t0 = tid * 4;                       // 4 points per thread
        const v4f xv = *(const v4f*)&xs[t0];
        const v4f yv = *(const v4f*)&ys[t0];
        float o[12];
#pragma unroll
        for (int j = 0; j < 4; ++j) {
            const float x = xv[j], y = yv[j];
            const float x2 = x * x, y2 = y * y, xy = x * y;
            const float nx = -(b0 + 2.f * b2 * x + b4 * y + 3.f * b5 * x2 + 2.f * b7 * xy + b8 * y2);
            const float ny = -(b1 + 2.f * b3 * y + b4 * x + 3.f * b6 * y2 + b7 * x2 + 2.f * b8 * xy);
            const float inv = 1.0f / sqrtf(nx * nx + ny * ny + 1.0f);
            o[3 * j + 0] = nx * inv;
            o[3 * j + 1] = ny * inv;
            o[3 * j + 2] = inv;
        }
        v4f* dst = (v4f*)(out_nn + ((size_t)b * NPTS + t0) * 3);   // 48B per thread, 16B aligned
#pragma unroll
        for (int q = 0; q < 3; ++q) {
            v4f pkt; pkt[0] = o[4 * q]; pkt[1] = o[4 * q + 1]; pkt[2] = o[4 * q + 2]; pkt[3] = o[4 * q + 3];
            __builtin_nontemporal_store(pkt, dst + q);
        }
    }
}

extern "C" void kernel_launch(void* const* d_in, const int* in_sizes, int n_in,
                              void* d_out, int out_size, void* d_ws, size_t ws_size,
                              hipStream_t stream) {
    const float* points  = (const float*)d_in[0];   // (B, 3, N) f32
    const float* weights = (const float*)d_in[1];   // (B, N) f32
    float* out = (float*)d_out;

    const int B = in_sizes[1] / NPTS;               // 8192
    float* out_beta = out;                          // (B, 10)
    float* out_nest = out + (size_t)B * 10;         // (B, 3)
    float* out_nn   = out + (size_t)B * 13;         // (B, N, 3)

    wjet_kernel<<<B, BLK, 0, stream>>>(points, weights, out_beta, out_nest, out_nn);
}